// CARAFE_23845658427768
// MI455X (gfx1250) — compile-verified
//
#include <hip/hip_runtime.h>

// CARAFE upsampling (K=5, G=1, scale=2) for MI455X / gfx1250.
//
// Roofline: features 33.5MB (+halo restage ~63MB via L2) + masks 13.1MB +
// output 134MB  ->  bandwidth-bound (~9us @ 23.3 TB/s). The 25-tap weighted
// reduction runs on the matrix pipe:
//
//   A (16 ch x 60 K)  : feature union-patch, K = dy*12 + wx over a 5-row x
//                       12-col window shared by 16 output cols & 2 output
//                       rows. 60 % 4 == 0 -> exactly 15 K-chunks, NO padding,
//                       no predicates in the compute loop. A is staged in LDS
//                       as sfeat[ch][60] so each chunk is ONE aligned
//                       ds_load_b64 with an immediate offset.
//   B (60 K x 16 col) : masks, zero outside each column's 5-tap window,
//                       built once per wave from LDS.
//   D (16 ch x 16 col): 64B-contiguous stores per half-wave.

typedef float v2f __attribute__((ext_vector_type(2)));
typedef float v8f __attribute__((ext_vector_type(8)));

#define CK2   25      // kernel taps
#define HF    128     // source H
#define WF    128     // source W
#define CH    256     // channels
#define HO    256     // output H
#define WO    256     // output W
#define KU    60      // union-patch K extent: 5 dy * 12 wx  (15 chunks of 4)
#define NCHK  15      // K chunks

// B-operand gather from LDS-staged masks.
// k = dy*12 + wx ; column jn -> source-col offset qj = jn>>1 ; dx = wx - qj.
__device__ __forceinline__ float bval(const float* sm, int i2, int k, int jn) {
  int dy = k / 12;
  int wx = k - dy * 12;
  int dx = wx - (jn >> 1);
  float v = 0.0f;
  if (dx >= 0 && dx < 5) {
    v = sm[(i2 * CK2 + dy * 5 + dx) * 16 + jn];
  }
  return v;
}

__global__ __launch_bounds__(256) void carafe_wmma_f32(
    const float* __restrict__ feat,   // (2, 256, 128, 128)
    const float* __restrict__ mask,   // (2, 25, 256, 256)
    float* __restrict__ out) {        // (2, 256, 256, 256)
  __shared__ float sfeat[CH * KU];        // [ch][k]  61,440 B
  __shared__ float smask[2 * CK2 * 16];   // [i2][tap][jn]  3,200 B

  const int n   = blockIdx.z;
  const int h   = blockIdx.y;           // source row
  const int w0  = blockIdx.x * 8;       // first source col of the 8-col tile
  const int tid = threadIdx.x;

  // ---- Stage masks for both output rows (coalesced 64B spans). ----
  for (int idx = tid; idx < 2 * CK2 * 16; idx += 256) {
    int i2  = idx / (CK2 * 16);
    int rem = idx - i2 * (CK2 * 16);
    int kk  = rem >> 4;
    int jn  = rem & 15;
    smask[idx] = mask[(((size_t)n * CK2 + kk) * HO + (2 * h + i2)) * WO
                      + (2 * w0 + jn)];
  }

  // ---- Stage the feature union-patch: sfeat[ch][k], k = dy*12 + wx.
  //      Thread t owns fixed k = t&63 (k<60 active), channel = (t>>6)+4*it.
  //      Per (ch,dy) the 12 wx are consecutive floats -> 48B spans. ----
  {
    const int k = tid & 63;
    if (k < KU) {
      const int dy  = k / 12;
      const int wx  = k - dy * 12;
      const int r   = h + dy - 2;
      const int c   = w0 + wx - 2;
      const bool ok = (r >= 0) && (r < HF) && (c >= 0) && (c < WF);
      const int ch0 = tid >> 6;
      const float* gp =
          feat + ((size_t)n * CH + ch0) * (HF * WF) + r * WF + c;
      float* lp = sfeat + ch0 * KU + k;
#pragma unroll 4
      for (int it = 0; it < 64; ++it) {   // 64 x 4 channels = 256
        float v = 0.0f;
        if (ok) v = gp[(size_t)it * 4 * HF * WF];
        lp[it * 4 * KU] = v;
      }
    }
  }
  __syncthreads();

  const int lane = tid & 31;
  const int wave = tid >> 5;
  const int ln16 = lane & 15;   // A row (channel-in-block) == B/D column (jn)
  const int half = lane >> 4;   // K sub-slot selector per ISA A/B layout

  // ---- Build B operands once per wave: 15 K-chunks x 2 output rows.
  //      v2f slot: .x = B[k0+2*half, jn], .y = B[k0+2*half+1, jn]. ----
  v2f B0[NCHK], B1[NCHK];
#pragma unroll
  for (int c = 0; c < NCHK; ++c) {
    int k0 = 4 * c + 2 * half;
    v2f b0, b1;
    b0.x = bval(smask, 0, k0,     ln16);
    b0.y = bval(smask, 0, k0 + 1, ln16);
    b1.x = bval(smask, 1, k0,     ln16);
    b1.y = bval(smask, 1, k0 + 1, ln16);
    B0[c] = b0;
    B1[c] = b1;
  }

  // ---- Each wave handles 2 channel blocks of 16 (8 waves x 2 = 256 ch). ----
  for (int blk = 0; blk < 2; ++blk) {
    const int c0 = (wave * 2 + blk) * 16;
    // A chunk base for this lane: sfeat[c0+ln16][2*half]; chunk c adds 16B.
    const float* ap = sfeat + (c0 + ln16) * KU + 2 * half;

    v8f d0 = {};
    v8f d1 = {};
#pragma unroll
    for (int c = 0; c < NCHK; ++c) {
      v2f a = *(const v2f*)(ap + 4 * c);   // one ds_load_b64, imm offset 16c
      d0 = __builtin_amdgcn_wmma_f32_16x16x4_f32(
          false, a, false, B0[c], (short)0, d0, false, false);
      d1 = __builtin_amdgcn_wmma_f32_16x16x4_f32(
          false, a, false, B1[c], (short)0, d1, false, false);
    }

    // ---- Store: lane holds D[M = v + 8*half, N = ln16]; each b32 store
    //      covers 16 consecutive output columns per half-wave. ----
    const int i0 = 2 * h;
    const int jg = 2 * w0 + ln16;
#pragma unroll
    for (int v = 0; v < 8; ++v) {
      int ch = c0 + v + 8 * half;
      size_t o = (((size_t)n * CH + ch) * HO + i0) * WO + jg;
      out[o]      = d0[v];
      out[o + WO] = d1[v];
    }
  }
}

extern "C" void kernel_launch(void* const* d_in, const int* in_sizes, int n_in,
                              void* d_out, int out_size, void* d_ws,
                              size_t ws_size, hipStream_t stream) {
  const float* feat = (const float*)d_in[0];   // (2,256,128,128) fp32
  const float* mask = (const float*)d_in[1];   // (2,25,256,256)  fp32
  float* out = (float*)d_out;                  // (2,256,256,256) fp32

  dim3 grid(WF / 8, HF, 2);   // 16 x 128 x 2 = 4096 workgroups
  carafe_wmma_f32<<<grid, 256, 0, stream>>>(feat, mask, out);
}